// MoELayer_26190710571113
// MI455X (gfx1250) — compile-verified
//
#include <hip/hip_runtime.h>
#include <math.h>

typedef __attribute__((ext_vector_type(2))) float v2f;
typedef __attribute__((ext_vector_type(8))) float v8f;
typedef __attribute__((ext_vector_type(4))) unsigned int v4u;
typedef __attribute__((ext_vector_type(8))) int v8i;
typedef __attribute__((ext_vector_type(4))) int v4i;

#define Bn 32
#define Sn 32
#define Dn 768
#define FFn 3072
#define En 8
#define Kn 2

#define PROB_OFF (Bn * Sn * Dn)          // 786432
#define GATE_OFF (PROB_OFF + Bn * Kn)    // 786496

#define CHUNK 64                          // K-chunk staged per TDM transfer
#define LDS_STRIDE 66                     // 64 + 2 pad DWORDs (TDM pad feature)
#define CHUNK_LDS (Sn * LDS_STRIDE)       // floats per staging buffer (2112)

// ---------------------------------------------------------------------------
// TDM: DMA a 32-row x 64-col f32 tile (row stride = row_stride_elems) from
// global memory into LDS at lds_byte_off, inserting 2 pad DWORDs after every
// 64 DWORDs so the LDS row stride becomes 66 floats (bank-conflict-free for
// the WMMA A-layout reads). D# packed per CDNA5 ISA 8.3/8.4.
// ---------------------------------------------------------------------------
__device__ __forceinline__ void tdm_load_tile_2d(unsigned lds_byte_off,
                                                 const float* gsrc,
                                                 unsigned long long row_stride_elems) {
  const unsigned long long ga = (unsigned long long)(uintptr_t)gsrc;
  v4u g0;
  g0.x = 1u;                                                  // count=1 (valid), user mode
  g0.y = lds_byte_off;                                        // lds_addr
  g0.z = (unsigned)ga;                                        // global_addr[31:0]
  g0.w = (unsigned)((ga >> 32) & 0x1FFFFFFull) | (2u << 30);  // global_addr[56:32] | type=2

  v8i g1;
  g1[0] = (int)((2u << 16) |    // data_size = 4 bytes
                (1u << 20) |    // pad_enable
                (5u << 22) |    // pad_interval: 64 DWORDs
                (1u << 25));    // pad_amount: 2 DWORDs
  // tensor_dim0 = tensor_dim1 = 1<<20 (tiles always in bounds; no OOB clip)
  g1[1] = 0;                                            // bar_addr=0, tensor_dim0[15:0]=0
  g1[2] = (int)(0x10u);                                 // tensor_dim0[31:16]=0x10, td1[15:0]=0
  g1[3] = (int)(0x10u | ((unsigned)CHUNK << 16));       // td1[31:16]=0x10, tile_dim0=64
  g1[4] = Sn;                                           // tile_dim1=32, tile_dim2=0
  g1[5] = (int)(unsigned)(row_stride_elems & 0xFFFFFFFFull);   // dim0_stride[31:0]
  g1[6] = (int)(unsigned)((row_stride_elems >> 32) & 0xFFFFull); // dim0_stride[47:32]
  g1[7] = 0;                                            // dim1_stride = 0 (2D tile)
  const v4i gz4 = {0, 0, 0, 0};
  const v8i gz8 = {0, 0, 0, 0, 0, 0, 0, 0};
  __builtin_amdgcn_tensor_load_to_lds(g0, g1, gz4, gz4, gz8, 0);
}

// ---------------------------------------------------------------------------
// Kernel 1: sentence-mean gating, softmax, top-2. One block per sentence.
// ---------------------------------------------------------------------------
__global__ __launch_bounds__(256) void moe_gate_kernel(
    const float* __restrict__ x, const float* __restrict__ Wg,
    const float* __restrict__ bg, float* __restrict__ out,
    int* __restrict__ gate_ws) {
  __shared__ float xa[Dn];
  __shared__ float lg[En];
  const int b = blockIdx.x;
  const int t = threadIdx.x;

  for (int d = t; d < Dn; d += 256) {
    float s = 0.f;
#pragma unroll 4
    for (int ss = 0; ss < Sn; ++ss)
      s += x[(size_t)(b * Sn + ss) * Dn + d];
    xa[d] = s * (1.0f / (float)Sn);
  }
  __syncthreads();

  if (t < En) {
    float acc = bg[t];
    for (int d = 0; d < Dn; ++d) acc += xa[d] * Wg[d * En + t];
    lg[t] = acc;
  }
  __syncthreads();

  if (t == 0) {
    float v[En];
#pragma unroll
    for (int i = 0; i < En; ++i) v[i] = lg[i];
    int i0 = 0;
#pragma unroll
    for (int i = 1; i < En; ++i)
      if (v[i] > v[i0]) i0 = i;
    int i1 = (i0 == 0) ? 1 : 0;
#pragma unroll
    for (int i = 0; i < En; ++i)
      if (i != i0 && v[i] > v[i1]) i1 = i;
    float mx = v[i0], den = 0.f;
#pragma unroll
    for (int i = 0; i < En; ++i) den += expf(v[i] - mx);
    const float inv = 1.0f / den;
    out[PROB_OFF + b * Kn + 0] = expf(v[i0] - mx) * inv;
    out[PROB_OFF + b * Kn + 1] = expf(v[i1] - mx) * inv;
    out[GATE_OFF + b * Kn + 0] = (float)i0;
    out[GATE_OFF + b * Kn + 1] = (float)i1;
    gate_ws[b * Kn + 0] = i0;
    gate_ws[b * Kn + 1] = i1;
  }
}

// ---------------------------------------------------------------------------
// Kernel 2: up-projection + exact GELU for each selected (sentence, expert)
// pair. 32 x 128 output tile per block; K=768 via f32 WMMA 16x16x4.
// A tiles double-buffered into LDS by the Tensor Data Mover (wave 0 issues,
// s_wait_tensorcnt + workgroup barrier for hand-off). grid=(FFn/128, Bn*Kn)
// ---------------------------------------------------------------------------
__global__ __launch_bounds__(256) void moe_up_kernel(
    const float* __restrict__ x, const float* __restrict__ W1,
    const float* __restrict__ b1, const int* __restrict__ gate_ws,
    float* __restrict__ hbuf) {
  __shared__ float xs[2 * CHUNK_LDS];
  const int pair = blockIdx.y;
  const int b = pair >> 1;
  const int e = gate_ws[pair];
  const int nbase = blockIdx.x * 128;
  const int t = threadIdx.x;
  const int wv = t >> 5;
  const int lane = t & 31;
  const int lo = lane & 15;
  const int hi = lane >> 4;
  const int ncol = nbase + wv * 16 + lo;
  const float* W1e = W1 + (size_t)e * Dn * FFn;
  const float* xrow = x + (size_t)b * Sn * Dn;
  const unsigned lds_base = (unsigned)(uintptr_t)(void*)xs;  // LDS byte offset

  v8f c0 = {0.f, 0.f, 0.f, 0.f, 0.f, 0.f, 0.f, 0.f};
  v8f c1 = {0.f, 0.f, 0.f, 0.f, 0.f, 0.f, 0.f, 0.f};

  const int nch = Dn / CHUNK;  // 12
  if (wv == 0) tdm_load_tile_2d(lds_base, xrow, Dn);
  int buf = 0;
  for (int c = 0; c < nch; ++c) {
    __syncthreads();  // everyone done reading the buffer we are about to DMA
    if (wv == 0) {
      if (c + 1 < nch) {
        tdm_load_tile_2d(lds_base + (unsigned)((buf ^ 1) * CHUNK_LDS * 4),
                         xrow + (size_t)(c + 1) * CHUNK, Dn);
        __builtin_amdgcn_s_wait_tensorcnt(1);  // chunk c done, c+1 in flight
      } else {
        __builtin_amdgcn_s_wait_tensorcnt(0);  // last chunk done
      }
    }
    __syncthreads();  // chunk c resident in LDS for all waves
    const float* xc = &xs[buf * CHUNK_LDS];
    const float* bb0 = W1e + (size_t)(c * CHUNK) * FFn + ncol;
#pragma unroll
    for (int kc = 0; kc < CHUNK; kc += 4) {
      const int kr = kc + 2 * hi;  // this lane's K pair: kr, kr+1
      const v2f a0 = *(const v2f*)&xc[lo * LDS_STRIDE + kr];
      const v2f a1 = *(const v2f*)&xc[(lo + 16) * LDS_STRIDE + kr];
      const float* bp = bb0 + (size_t)kr * FFn;
      v2f bbv;
      bbv.x = bp[0];
      bbv.y = bp[FFn];
      c0 = __builtin_amdgcn_wmma_f32_16x16x4_f32(false, a0, false, bbv,
                                                 (short)0, c0, false, false);
      c1 = __builtin_amdgcn_wmma_f32_16x16x4_f32(false, a1, false, bbv,
                                                 (short)0, c1, false, false);
    }
    buf ^= 1;
  }

  const float bias = b1[(size_t)e * FFn + ncol];
  float* hp = hbuf + (size_t)pair * Sn * FFn;
#pragma unroll
  for (int r = 0; r < 8; ++r) {
    const int m0 = r + 8 * hi;  // C layout: M = r + 8*(lane>=16), N = lane&15
    float v0c = c0[r] + bias;
    v0c = 0.5f * v0c * (1.0f + erff(v0c * 0.70710678118654752f));
    hp[(size_t)m0 * FFn + ncol] = v0c;
    float v1c = c1[r] + bias;
    v1c = 0.5f * v1c * (1.0f + erff(v1c * 0.70710678118654752f));
    hp[(size_t)(m0 + 16) * FFn + ncol] = v1c;
  }
}

// ---------------------------------------------------------------------------
// Kernel 3: down-projection + unweighted combine of the K=2 selected experts.
// Flattened 96-chunk K loop (2 experts x FFn/64) with the same TDM
// double-buffered staging; both experts accumulate into one WMMA tile.
// grid = (Dn/128, Bn)
// ---------------------------------------------------------------------------
__global__ __launch_bounds__(256) void moe_down_kernel(
    const float* __restrict__ hbuf, const float* __restrict__ W2,
    const float* __restrict__ b2, const int* __restrict__ gate_ws,
    float* __restrict__ out) {
  __shared__ float hs[2 * CHUNK_LDS];
  const int b = blockIdx.y;
  const int nbase = blockIdx.x * 128;
  const int t = threadIdx.x;
  const int wv = t >> 5;
  const int lane = t & 31;
  const int lo = lane & 15;
  const int hi = lane >> 4;
  const int ncol = nbase + wv * 16 + lo;

  const int e0 = gate_ws[b * Kn + 0];
  const int e1 = gate_ws[b * Kn + 1];
  const float* hp0 = hbuf + (size_t)(b * Kn + 0) * Sn * FFn;
  const float* hp1 = hbuf + (size_t)(b * Kn + 1) * Sn * FFn;
  const float* W2e0 = W2 + (size_t)e0 * FFn * Dn;
  const float* W2e1 = W2 + (size_t)e1 * FFn * Dn;
  const unsigned lds_base = (unsigned)(uintptr_t)(void*)hs;

  v8f c0 = {0.f, 0.f, 0.f, 0.f, 0.f, 0.f, 0.f, 0.f};
  v8f c1 = {0.f, 0.f, 0.f, 0.f, 0.f, 0.f, 0.f, 0.f};

  const int half = FFn / CHUNK;  // 48 chunks per expert
  const int nch = 2 * half;      // 96 total

  if (wv == 0) tdm_load_tile_2d(lds_base, hp0, FFn);
  int buf = 0;
  for (int c = 0; c < nch; ++c) {
    __syncthreads();
    if (wv == 0) {
      if (c + 1 < nch) {
        const int cn = c + 1;
        const float* nsrc = (cn < half)
                                ? hp0 + (size_t)cn * CHUNK
                                : hp1 + (size_t)(cn - half) * CHUNK;
        tdm_load_tile_2d(lds_base + (unsigned)((buf ^ 1) * CHUNK_LDS * 4),
                         nsrc, FFn);
        __builtin_amdgcn_s_wait_tensorcnt(1);
      } else {
        __builtin_amdgcn_s_wait_tensorcnt(0);
      }
    }
    __syncthreads();
    const float* hc = &hs[buf * CHUNK_LDS];
    const float* bb0 = (c < half)
                           ? W2e0 + (size_t)(c * CHUNK) * Dn + ncol
                           : W2e1 + (size_t)((c - half) * CHUNK) * Dn + ncol;
#pragma unroll
    for (int kc = 0; kc < CHUNK; kc += 4) {
      const int kr = kc + 2 * hi;
      const v2f a0 = *(const v2f*)&hc[lo * LDS_STRIDE + kr];
      const v2f a1 = *(const v2f*)&hc[(lo + 16) * LDS_STRIDE + kr];
      const float* bp = bb0 + (size_t)kr * Dn;
      v2f bbv;
      bbv.x = bp[0];
      bbv.y = bp[Dn];
      c0 = __builtin_amdgcn_wmma_f32_16x16x4_f32(false, a0, false, bbv,
                                                 (short)0, c0, false, false);
      c1 = __builtin_amdgcn_wmma_f32_16x16x4_f32(false, a1, false, bbv,
                                                 (short)0, c1, false, false);
    }
    buf ^= 1;
  }

  const float bias = b2[(size_t)e0 * Dn + ncol] + b2[(size_t)e1 * Dn + ncol];
#pragma unroll
  for (int r = 0; r < 8; ++r) {
    const int m0 = r + 8 * hi;
    out[(size_t)(b * Sn + m0) * Dn + ncol] = c0[r] + bias;
    out[(size_t)(b * Sn + m0 + 16) * Dn + ncol] = c1[r] + bias;
  }
}

// ---------------------------------------------------------------------------
extern "C" void kernel_launch(void* const* d_in, const int* in_sizes, int n_in,
                              void* d_out, int out_size, void* d_ws,
                              size_t ws_size, hipStream_t stream) {
  const float* x = (const float*)d_in[0];
  // d_in[1] attention_mask: reference overwrites it with ones -> unused
  const float* Wg = (const float*)d_in[2];
  const float* bg = (const float*)d_in[3];
  const float* W1 = (const float*)d_in[4];
  const float* b1 = (const float*)d_in[5];
  const float* W2 = (const float*)d_in[6];
  const float* b2 = (const float*)d_in[7];
  float* out = (float*)d_out;

  int* gate_ws = (int*)d_ws;                   // 64 ints
  float* hbuf = (float*)((char*)d_ws + 1024);  // 64*32*3072 f32 = 25 MB

  moe_gate_kernel<<<dim3(Bn), 256, 0, stream>>>(x, Wg, bg, out, gate_ws);
  moe_up_kernel<<<dim3(FFn / 128, Bn * Kn), 256, 0, stream>>>(x, W1, b1,
                                                              gate_ws, hbuf);
  moe_down_kernel<<<dim3(Dn / 128, Bn), 256, 0, stream>>>(hbuf, W2, b2,
                                                          gate_ws, out);
}